// ddpOptimizer_49349174231142
// MI455X (gfx1250) — compile-verified
//
#include <hip/hip_runtime.h>

#define NS 16
#define NC 8
#define NSC 24

typedef float v2f __attribute__((ext_vector_type(2)));
typedef float v8f __attribute__((ext_vector_type(8)));

// ---------- CDNA5 async global->LDS copy (ASYNCcnt-tracked) ----------
__device__ __forceinline__ void async_b128(unsigned lds_off, const float* g) {
  asm volatile("global_load_async_to_lds_b128 %0, %1, off"
               :: "v"(lds_off), "v"((unsigned long long)(uintptr_t)g)
               : "memory");
}
__device__ __forceinline__ void wait_async0() {
  asm volatile("s_wait_asynccnt 0x0" ::: "memory");
}

// Issue Q_t (24x24 f32, row-major) into a padded 32-stride LDS tile.
// 144 float4 groups -> 5 wave-wide issues; overflow lanes hit a dummy slot.
__device__ __forceinline__ void issue_q(float (*dst)[32], const float* src,
                                        unsigned dummyOff, int lane) {
#pragma unroll
  for (int i = 0; i < 5; ++i) {
    const int g  = i * 32 + lane;
    const int gc = (g < 144) ? g : 143;
    const int r  = gc / 6, c4 = gc - r * 6;
    const unsigned off =
        (g < 144) ? (unsigned)(uintptr_t)&dst[r][c4 * 4] : dummyOff;
    async_b128(off, src + gc * 4);
  }
}
// Issue F_t (16x24 f32) into padded 32-stride LDS tile: exactly 96 groups.
__device__ __forceinline__ void issue_f(float (*dst)[32], const float* src,
                                        int lane) {
#pragma unroll
  for (int i = 0; i < 3; ++i) {
    const int g = i * 32 + lane;
    const int r = g / 6, c4 = g - r * 6;
    async_b128((unsigned)(uintptr_t)&dst[r][c4 * 4], src + g * 4);
  }
}

// Accumulate one 16x16 f32 tile: acc += A(16xKD) * B(KDx16), data in LDS.
// TA: A stored transposed (physical [k][m]); TB: B stored transposed ([n][k]).
// A-frag layout (ISA 7.12.2): lane L holds row M=L&15; vgpr j holds K = 2*(L>=16)+j.
// C/D layout: vgpr r holds D[r + 8*(L>=16)][L&15].
template <bool TA, bool TB, int KD>
__device__ __forceinline__ void wmma_acc_16x16(const float* __restrict__ A, int lda,
                                               const float* __restrict__ B, int ldb,
                                               v8f& acc, int lane) {
  const int half = lane >> 4;
  const int l16  = lane & 15;
#pragma unroll
  for (int k0 = 0; k0 < KD; k0 += 4) {
    v2f a, b;
#pragma unroll
    for (int j = 0; j < 2; ++j) {
      const int k = k0 + 2 * half + j;
      a[j] = TA ? A[k * lda + l16] : A[l16 * lda + k];
      b[j] = TB ? B[l16 * ldb + k] : B[k * ldb + l16];
    }
    acc = __builtin_amdgcn_wmma_f32_16x16x4_f32(false, a, false, b, (short)0, acc,
                                                false, false);
  }
}

// ---------------- Backward Riccati sweep: one wave per batch chain ----------------
__global__ __launch_bounds__(32) void lqr_bwd_kernel(
    const float* __restrict__ Q, const float* __restrict__ p,
    const float* __restrict__ F, const float* __restrict__ c1,
    float* __restrict__ Kws, float* __restrict__ kws, int T) {
  __shared__ __align__(16) float sQbuf[2][32][32]; // Q_t then Qh, double-buffered
  __shared__ __align__(16) float sFbuf[2][16][32]; // F_t, double-buffered
  __shared__ __align__(16) float sDummy[4];        // async overflow target
  __shared__ float sV[16][16];   // value matrix V
  __shared__ float sv[16];       // value vector v
  __shared__ float sW[16][32];   // W = V @ F
  __shared__ float sqh[32];      // qh (24 used)
  __shared__ float sc[16];       // c_t
  __shared__ float st[16];       // V c + v
  __shared__ float sS[8][26];    // augmented solve [Quu | Qux | qu]
  __shared__ float sK[16][16];   // K (8x16), rows 8..15 zero
  __shared__ float sk[8];        // k
  __shared__ float sKtQ[16][16]; // K^T Quu (16x8), cols 8..15 zero
  __shared__ float sVn[16][16];  // un-symmetrized Vn

  const int b    = blockIdx.x;
  const int lane = threadIdx.x;
  const int half = lane >> 4;
  const int l16  = lane & 15;
  const unsigned dummyOff = (unsigned)(uintptr_t)&sDummy[0];

  for (int idx = lane; idx < 256; idx += 32) {
    sV[idx >> 4][idx & 15] = 0.0f;
    sK[idx >> 4][idx & 15] = 0.0f;  // pad rows of K stay zero forever
  }
  // Zero both padded buffers once; pads are re-zeroed by WMMA stores (Q) or
  // never touched again (F), so the zero-pad invariant holds across steps.
  for (int idx = lane; idx < 2 * 32 * 32; idx += 32)
    sQbuf[0][0][idx] = 0.0f;
  for (int idx = lane; idx < 2 * 16 * 32; idx += 32)
    sFbuf[0][0][idx] = 0.0f;
  if (lane < 16) sv[lane] = 0.0f;
  __syncthreads();

  // Prologue: async-load tiles for t = T-1 into buffer 0.
  {
    const size_t bt0 = (size_t)b * T + (T - 1);
    issue_q(sQbuf[0], Q + bt0 * (NSC * NSC), dummyOff, lane);
    issue_f(sFbuf[0], F + bt0 * (NS * NSC), lane);
  }
  wait_async0();
  __syncthreads();

  int cur = 0;
  for (int t = T - 1; t >= 0; --t) {
    const size_t bt = (size_t)b * T + t;
    const float* pg = p + bt * NSC;
    const float* cg = c1 + bt * NS;
    float (*Qc)[32] = sQbuf[cur];
    float (*Fc)[32] = sFbuf[cur];

    // Kick off next step's tile loads into the other buffer (overlapped with
    // this step's compute; completion enforced at the loop tail).
    if (t > 0) {
      issue_q(sQbuf[cur ^ 1], Q + (bt - 1) * (NSC * NSC), dummyOff, lane);
      issue_f(sFbuf[cur ^ 1], F + (bt - 1) * (NS * NSC), lane);
    }
    // Warm L2 two steps ahead for the async engine; one step for p/c.
    if (t > 1) {
      const char* Qn = (const char*)(Q + (bt - 2) * (NSC * NSC));
      const char* Fn = (const char*)(F + (bt - 2) * (NS * NSC));
      __builtin_prefetch(Qn + lane * 128, 0, 3);
      __builtin_prefetch(Fn + lane * 64, 0, 3);
    }
    if (t > 0 && lane == 0) {
      __builtin_prefetch(pg - NSC, 0, 3);
      __builtin_prefetch(cg - NS, 0, 3);
    }

    if (lane < 16) sc[lane] = cg[lane];
    __syncthreads();

    // W = V @ F   (16 x 32, two 16-wide tiles)
#pragma unroll
    for (int n0 = 0; n0 < 32; n0 += 16) {
      v8f acc = {};
      wmma_acc_16x16<false, false, 16>(&sV[0][0], 16, &Fc[0][n0], 32, acc, lane);
#pragma unroll
      for (int r = 0; r < 8; ++r) sW[r + 8 * half][n0 + l16] = acc[r];
    }
    __syncthreads();

    // Qh = Q_t + F^T @ W : C initialized from Qc (holds Q_t), result overwrites
    // the same tile. Per-lane read/write sets are identical -> no extra syncs.
#pragma unroll
    for (int m0 = 0; m0 < 32; m0 += 16) {
#pragma unroll
      for (int n0 = 0; n0 < 32; n0 += 16) {
        v8f acc;
#pragma unroll
        for (int r = 0; r < 8; ++r) acc[r] = Qc[m0 + r + 8 * half][n0 + l16];
        wmma_acc_16x16<true, false, 16>(&Fc[0][m0], 32, &sW[0][n0], 32, acc, lane);
#pragma unroll
        for (int r = 0; r < 8; ++r) Qc[m0 + r + 8 * half][n0 + l16] = acc[r];
      }
    }
    __syncthreads();

    // st = V c + v ;  qh = p + F^T st
    if (lane < 16) {
      float s = sv[lane];
      for (int j = 0; j < 16; ++j) s += sV[lane][j] * sc[j];
      st[lane] = s;
    }
    __syncthreads();
    if (lane < NSC) {
      float s = pg[lane];
      for (int kx = 0; kx < 16; ++kx) s += Fc[kx][lane] * st[kx];
      sqh[lane] = s;
    }
    __syncthreads();

    // Build augmented system [Quu | Qux | qu]  (8 x 25)
    for (int idx = lane; idx < 8 * 25; idx += 32) {
      const int r = idx / 25, j = idx % 25;
      float val;
      if (j < 8)       val = Qc[16 + r][16 + j];
      else if (j < 24) val = Qc[16 + r][j - 8];
      else             val = sqh[16 + r];
      sS[r][j] = val;
    }
    __syncthreads();

    // Gauss-Jordan elimination (Quu is SPD -> no pivoting). Lane j owns column j.
    for (int kk = 0; kk < 8; ++kk) {
      const float rp = 1.0f / sS[kk][kk];
      float fz[8];
#pragma unroll
      for (int r = 0; r < 8; ++r) fz[r] = sS[r][kk] * rp;
      __syncthreads();
      if (lane < 25) {
        const float pvt = sS[kk][lane];
#pragma unroll
        for (int r = 0; r < 8; ++r)
          if (r != kk) sS[r][lane] -= fz[r] * pvt;
      }
      __syncthreads();
    }
    // K = -Quu^{-1} Qux ; k = -Quu^{-1} qu
    if (lane < 17) {
      for (int cdx = 0; cdx < 8; ++cdx) {
        const float x = -sS[cdx][8 + lane] / sS[cdx][cdx];
        if (lane < 16) sK[cdx][lane] = x;
        else           sk[cdx] = x;
      }
    }
    __syncthreads();

    // KtQ = K^T @ Quu   (16x8, padded; Quu read out of padded Qc)
    {
      v8f acc = {};
      wmma_acc_16x16<true, false, 8>(&sK[0][0], 16, &Qc[16][16], 32, acc, lane);
#pragma unroll
      for (int r = 0; r < 8; ++r) sKtQ[r + 8 * half][l16] = acc[r];
    }
    __syncthreads();

    // Vn = Qxx + Qxu@K + K^T@Qux + KtQ@K
    {
      v8f acc;
#pragma unroll
      for (int r = 0; r < 8; ++r) acc[r] = Qc[r + 8 * half][l16];  // Qxx
      wmma_acc_16x16<false, false, 8>(&Qc[0][16], 32, &sK[0][0], 16, acc, lane);
      wmma_acc_16x16<true,  false, 8>(&sK[0][0], 16, &Qc[16][0], 32, acc, lane);
      wmma_acc_16x16<false, false, 8>(&sKtQ[0][0], 16, &sK[0][0], 16, acc, lane);
#pragma unroll
      for (int r = 0; r < 8; ++r) sVn[r + 8 * half][l16] = acc[r];
    }
    __syncthreads();

    // V = 0.5 (Vn + Vn^T)
#pragma unroll
    for (int r = 0; r < 8; ++r) {
      const int i = r + 8 * half, j = l16;
      sV[i][j] = 0.5f * (sVn[i][j] + sVn[j][i]);
    }
    // v = qx + Qxu k + K^T qu + KtQ k
    if (lane < 16) {
      float s = sqh[lane];
      for (int cdx = 0; cdx < 8; ++cdx) {
        s += Qc[lane][16 + cdx] * sk[cdx];
        s += sK[cdx][lane] * sqh[16 + cdx];
        s += sKtQ[lane][cdx] * sk[cdx];
      }
      sv[lane] = s;
    }
    // Persist gains for the forward pass.
    for (int idx = lane; idx < NC * NS; idx += 32) {
      const int cdx = idx >> 4, i = idx & 15;
      Kws[(bt * NC + cdx) * NS + i] = sK[cdx][i];
    }
    if (lane < NC) kws[bt * NC + lane] = sk[lane];

    // Next iteration's tiles must be resident before we switch buffers.
    wait_async0();
    __syncthreads();
    cur ^= 1;
  }
}

// ---------------- Forward rollout: one wave per batch chain ----------------
__global__ __launch_bounds__(32) void lqr_fwd_kernel(
    const float* __restrict__ F, const float* __restrict__ c1,
    const float* __restrict__ x0, const float* __restrict__ Kws,
    const float* __restrict__ kws, float* __restrict__ out, int T) {
  __shared__ float sx[NS];
  __shared__ float stau[NSC];
  __shared__ float sK[NC][NS];

  const int b    = blockIdx.x;
  const int lane = threadIdx.x;

  if (lane < NS) sx[lane] = x0[(size_t)b * NS + lane];
  __syncthreads();

  for (int t = 0; t < T; ++t) {
    const size_t bt = (size_t)b * T + t;

    if (t + 1 < T) {  // prefetch next step's gains and dynamics
      const char* Kn = (const char*)(Kws + (bt + 1) * (NC * NS));
      const char* Fn = (const char*)(F + (bt + 1) * (NS * NSC));
      __builtin_prefetch(Kn + lane * 32, 0, 3);
      __builtin_prefetch(Fn + lane * 64, 0, 3);
    }

    for (int idx = lane; idx < NC * NS; idx += 32)
      sK[idx >> 4][idx & 15] = Kws[bt * (NC * NS) + idx];
    __syncthreads();

    if (lane < NSC) {
      float v;
      if (lane < NS) {
        v = sx[lane];
      } else {
        const int cdx = lane - NS;
        float s = kws[bt * NC + cdx];
        for (int i = 0; i < NS; ++i) s += sK[cdx][i] * sx[i];
        v = s;
      }
      stau[lane] = v;
      out[bt * NSC + lane] = v;  // tau_t
    }
    __syncthreads();

    if (lane < NS) {
      const float* Fg = F + bt * (NS * NSC);
      float s = c1[bt * NS + lane];
      for (int j = 0; j < NSC; ++j) s += Fg[lane * NSC + j] * stau[j];
      sx[lane] = s;  // x_{t+1}
    }
    __syncthreads();
  }
}

extern "C" void kernel_launch(void* const* d_in, const int* in_sizes, int n_in,
                              void* d_out, int out_size, void* d_ws, size_t ws_size,
                              hipStream_t stream) {
  (void)n_in; (void)out_size; (void)ws_size;
  const float* Q  = (const float*)d_in[0];
  const float* p  = (const float*)d_in[1];
  const float* F  = (const float*)d_in[2];
  const float* c1 = (const float*)d_in[3];
  const float* x0 = (const float*)d_in[4];
  float* out = (float*)d_out;

  const int B = in_sizes[4] / NS;                // x_init: (B, 16)
  const int T = in_sizes[3] / (B * NS);          // c1:     (B, T, 16)

  float* Kws = (float*)d_ws;                     // (B,T,8,16)
  float* kws = Kws + (size_t)B * T * NC * NS;    // (B,T,8)

  lqr_bwd_kernel<<<B, 32, 0, stream>>>(Q, p, F, c1, Kws, kws, T);
  lqr_fwd_kernel<<<B, 32, 0, stream>>>(F, c1, x0, Kws, kws, out, T);
}